// GCN_84267258347588
// MI455X (gfx1250) — compile-verified
//
#include <hip/hip_runtime.h>
#include <cstdint>

// ---------------------------------------------------------------------------
// GCN (3x GCNConv + mean-pool + linear) for MI455X / gfx1250, wave32.
//
// - Dense layers: exact-fp32 V_WMMA_F32_16X16X4_F32. One wave computes a
//   16x64 output strip (4 accumulators share each A fragment -> 4x less
//   A traffic, back-to-back WMMAs pipeline in the matrix unit).
// - Aggregation: one-time CSR-by-dst build (count + scan + fill), then
//   per-layer pure-gather aggregation: one wave per node, 2 features/lane,
//   coalesced 256B/edge reads from the L2-resident t table. No per-layer
//   atomics, no zero-fill, bias+ReLU fused. (Working set ~90MB << 192MB L2.)
// ---------------------------------------------------------------------------

typedef __attribute__((ext_vector_type(2))) float v2f;
typedef __attribute__((ext_vector_type(8))) float v8f;

#define N_NODES 100000
#define N_EDGES 1600000
#define F_IN    128
#define H_DIM   64
#define G_NUM   128
#define C_CLS   10
#define SCAN_B  256

// ---------------- utility kernels ----------------

__global__ void fill_f32(float* __restrict__ p, float v, int n) {
    int i = blockIdx.x * blockDim.x + threadIdx.x;
    if (i < n) p[i] = v;
}

__global__ void fill_i32(int* __restrict__ p, int v, int n) {
    int i = blockIdx.x * blockDim.x + threadIdx.x;
    if (i < n) p[i] = v;
}

// ---------------- degree / norm ----------------

__global__ void deg_edges(const int* __restrict__ dst, float* __restrict__ deg, int E) {
    int e = blockIdx.x * blockDim.x + threadIdx.x;
    if (e < E) atomicAdd(&deg[dst[e]], 1.0f);
}

__global__ void dinv_kernel(float* __restrict__ d, int n) {
    int i = blockIdx.x * blockDim.x + threadIdx.x;
    if (i < n) d[i] = rsqrtf(d[i]);   // deg >= 1 (self loop included)
}

__global__ void norm_kernel(const int* __restrict__ src, const int* __restrict__ dst,
                            const float* __restrict__ dinv, float* __restrict__ norm,
                            int E, int Nn) {
    int e = blockIdx.x * blockDim.x + threadIdx.x;
    if (e >= E + Nn) return;
    float v;
    if (e < E) {
        v = dinv[src[e]] * dinv[dst[e]];
    } else {
        float d = dinv[e - E];
        v = d * d;
    }
    norm[e] = v;
}

// ---------------- CSR-by-dst build (one-time counting sort) ----------------

__global__ void count_edges(const int* __restrict__ dst, int* __restrict__ cnt,
                            int E, int Nn) {
    int e = blockIdx.x * blockDim.x + threadIdx.x;
    if (e >= E + Nn) return;
    int d = (e < E) ? dst[e] : (e - E);
    atomicAdd(&cnt[d], 1);
}

// block-level exclusive scan (Hillis-Steele in LDS) + block totals
__global__ void scan_block(const int* __restrict__ in, int* __restrict__ out_excl,
                           int* __restrict__ blockSums, int n) {
    __shared__ int sh[SCAN_B];
    int tid = threadIdx.x;
    int i = blockIdx.x * SCAN_B + tid;
    int v = (i < n) ? in[i] : 0;
    sh[tid] = v;
    __syncthreads();
    for (int off = 1; off < SCAN_B; off <<= 1) {
        int t = (tid >= off) ? sh[tid - off] : 0;
        __syncthreads();
        sh[tid] += t;
        __syncthreads();
    }
    if (i < n) out_excl[i] = sh[tid] - v;          // exclusive
    if (tid == SCAN_B - 1) blockSums[blockIdx.x] = sh[SCAN_B - 1];
}

__global__ void scan_sums(int* __restrict__ blockSums, int nb) {
    if (blockIdx.x == 0 && threadIdx.x == 0) {     // nb ~ 391: trivial serial
        int acc = 0;
        for (int i = 0; i < nb; ++i) { int v = blockSums[i]; blockSums[i] = acc; acc += v; }
    }
}

__global__ void scan_add(int* __restrict__ out_excl, const int* __restrict__ blockSums,
                         int n) {
    int i = blockIdx.x * SCAN_B + threadIdx.x;
    if (i < n) out_excl[i] += blockSums[blockIdx.x];
}

__global__ void fill_csr(const int* __restrict__ dst, const int* __restrict__ row_start,
                         int* __restrict__ cursor, int* __restrict__ eid, int E, int Nn) {
    int e = blockIdx.x * blockDim.x + threadIdx.x;
    if (e >= E + Nn) return;
    int d = (e < E) ? dst[e] : (e - E);
    int slot = atomicAdd(&cursor[d], 1);
    eid[row_start[d] + slot] = e;
}

// ---------------- fp32 WMMA GEMM:  out[M,64] = A[M,K] @ W[K,64] ----------------
// One wave per 16-row strip; 4 N-tiles share each A fragment; K step = 4.

__global__ void gemm_wmma_strip(const float* __restrict__ A, const float* __restrict__ W,
                                float* __restrict__ out, int M, int K) {
    const int lane = threadIdx.x & 31;
    const int wave = threadIdx.x >> 5;
    const int tm   = blockIdx.x * (blockDim.x >> 5) + wave;
    if (tm >= M / 16) return;                      // wave-uniform

    const int r    = lane & 15;
    const int half = lane >> 4;
    const int koff = half << 1;                    // 0 or 2

    const float* Arow = A + (size_t)(tm * 16 + r) * K;

    v8f acc0 = {}, acc1 = {}, acc2 = {}, acc3 = {};
    for (int k = 0; k < K; k += 4) {
        const float2 av = *(const float2*)(Arow + k + koff);
        v2f a; a.x = av.x; a.y = av.y;
        const float* w0 = W + (k + koff)     * H_DIM + r;
        const float* w1 = W + (k + koff + 1) * H_DIM + r;
        v2f b0, b1, b2, b3;
        b0.x = w0[0];  b0.y = w1[0];
        b1.x = w0[16]; b1.y = w1[16];
        b2.x = w0[32]; b2.y = w1[32];
        b3.x = w0[48]; b3.y = w1[48];
        acc0 = __builtin_amdgcn_wmma_f32_16x16x4_f32(false, a, false, b0, (short)0, acc0, false, false);
        acc1 = __builtin_amdgcn_wmma_f32_16x16x4_f32(false, a, false, b1, (short)0, acc1, false, false);
        acc2 = __builtin_amdgcn_wmma_f32_16x16x4_f32(false, a, false, b2, (short)0, acc2, false, false);
        acc3 = __builtin_amdgcn_wmma_f32_16x16x4_f32(false, a, false, b3, (short)0, acc3, false, false);
    }

    // D layout: VGPR i -> row = tm*16 + i + 8*half, col = r + {0,16,32,48}
    #pragma unroll
    for (int i = 0; i < 8; ++i) {
        float* o = out + (size_t)(tm * 16 + i + 8 * half) * H_DIM + r;
        o[0]  = acc0[i];
        o[16] = acc1[i];
        o[32] = acc2[i];
        o[48] = acc3[i];
    }
}

// ---------------- CSR gather aggregation (+ fused bias / ReLU) ----------------
// One wave per node; lane owns features [2*lane, 2*lane+1]; per edge the
// wave reads a coalesced 256B row of t. No atomics.

__global__ void aggregate_csr(const float* __restrict__ t, const int* __restrict__ src,
                              const float* __restrict__ norm,
                              const int* __restrict__ row_start,
                              const int* __restrict__ deg,
                              const int* __restrict__ eid,
                              const float* __restrict__ bias,
                              float* __restrict__ out,
                              int E, int Nn, int do_relu) {
    const int lane = threadIdx.x & 31;
    const int node = blockIdx.x * (blockDim.x >> 5) + (threadIdx.x >> 5);
    if (node >= Nn) return;                        // wave-uniform

    const int beg = row_start[node];
    const int nd  = deg[node];
    const int f   = lane << 1;

    float ax = 0.0f, ay = 0.0f;
    for (int j = 0; j < nd; ++j) {
        int   e = eid[beg + j];
        int   s = (e < E) ? src[e] : (e - E);
        float w = norm[e];
        const float2 v = *(const float2*)(t + (size_t)s * H_DIM + f);
        ax += v.x * w;
        ay += v.y * w;
    }
    float ox = ax + bias[f];
    float oy = ay + bias[f + 1];
    if (do_relu) { ox = fmaxf(ox, 0.0f); oy = fmaxf(oy, 0.0f); }
    float2 o; o.x = ox; o.y = oy;
    *(float2*)(out + (size_t)node * H_DIM + f) = o;
}

// ---------------- mean pool + readout ----------------

__global__ void pool_kernel(const float* __restrict__ h, const int* __restrict__ batch,
                            float* __restrict__ sums, float* __restrict__ cnt, int Nn) {
    int gid = blockIdx.x * blockDim.x + threadIdx.x;
    if (gid >= Nn * H_DIM) return;
    int nidx = gid >> 6;
    int f    = gid & (H_DIM - 1);
    int g    = batch[nidx];
    atomicAdd(&sums[g * H_DIM + f], h[gid]);
    if (f == 0) atomicAdd(&cnt[g], 1.0f);
}

__global__ void final_fc(const float* __restrict__ sums, const float* __restrict__ cnt,
                         const float* __restrict__ Wl, const float* __restrict__ bl,
                         float* __restrict__ out) {
    int gid = blockIdx.x * blockDim.x + threadIdx.x;
    if (gid >= G_NUM * C_CLS) return;
    int g = gid / C_CLS;
    int c = gid % C_CLS;
    float acc = 0.0f;
    #pragma unroll
    for (int hh = 0; hh < H_DIM; ++hh)
        acc += sums[g * H_DIM + hh] * Wl[hh * C_CLS + c];
    out[gid] = acc / fmaxf(cnt[g], 1.0f) + bl[c];
}

// ---------------- host-side orchestration ----------------

static inline int cdiv(int a, int b) { return (a + b - 1) / b; }

extern "C" void kernel_launch(void* const* d_in, const int* in_sizes, int n_in,
                              void* d_out, int out_size, void* d_ws, size_t ws_size,
                              hipStream_t stream) {
    (void)in_sizes; (void)n_in; (void)out_size; (void)ws_size;

    const float* x   = (const float*)d_in[0];
    const int*   ei  = (const int*)  d_in[1];   // [2, E]
    const int*   bat = (const int*)  d_in[2];   // [N]
    const float* W1  = (const float*)d_in[3];
    const float* b1  = (const float*)d_in[4];
    const float* W2  = (const float*)d_in[5];
    const float* b2  = (const float*)d_in[6];
    const float* W3  = (const float*)d_in[7];
    const float* b3  = (const float*)d_in[8];
    const float* Wl  = (const float*)d_in[9];
    const float* bl  = (const float*)d_in[10];
    float*       out = (float*)d_out;

    const int* src = ei;
    const int* dst = ei + N_EDGES;

    const int EN = N_EDGES + N_NODES;
    const int NH = N_NODES * H_DIM;
    const int NB = cdiv(N_NODES, SCAN_B);       // scan blocks (391)

    // ---- workspace layout (all 4-byte types; 16B-aligned segments) ----
    char* base = (char*)d_ws;
    float* dinv   = (float*)base;  base += (size_t)N_NODES * 4;
    float* norm   = (float*)base;  base += (size_t)EN * 4;
    float* t      = (float*)base;  base += (size_t)NH * 4;
    float* hA     = (float*)base;  base += (size_t)NH * 4;
    float* hB     = (float*)base;  base += (size_t)NH * 4;
    float* sums   = (float*)base;  base += (size_t)G_NUM * H_DIM * 4;
    float* cnt    = (float*)base;  base += (size_t)G_NUM * 4;
    int*   counts = (int*)base;    base += (size_t)N_NODES * 4;
    int*   rowst  = (int*)base;    base += (size_t)N_NODES * 4;
    int*   cursor = (int*)base;    base += (size_t)N_NODES * 4;
    int*   eid    = (int*)base;    base += (size_t)EN * 4;
    int*   bsums  = (int*)base;    base += (size_t)NB * 4;

    const int TB = 256;

    // ---- degree + symmetric norm (self loops included) ----
    fill_f32   <<<cdiv(N_NODES, TB), TB, 0, stream>>>(dinv, 1.0f, N_NODES);
    deg_edges  <<<cdiv(N_EDGES, TB), TB, 0, stream>>>(dst, dinv, N_EDGES);
    dinv_kernel<<<cdiv(N_NODES, TB), TB, 0, stream>>>(dinv, N_NODES);
    norm_kernel<<<cdiv(EN, TB), TB, 0, stream>>>(src, dst, dinv, norm, N_EDGES, N_NODES);

    // ---- one-time CSR-by-dst build ----
    fill_i32   <<<cdiv(N_NODES, TB), TB, 0, stream>>>(counts, 0, N_NODES);
    fill_i32   <<<cdiv(N_NODES, TB), TB, 0, stream>>>(cursor, 0, N_NODES);
    count_edges<<<cdiv(EN, TB), TB, 0, stream>>>(dst, counts, N_EDGES, N_NODES);
    scan_block <<<NB, SCAN_B, 0, stream>>>(counts, rowst, bsums, N_NODES);
    scan_sums  <<<1, 32, 0, stream>>>(bsums, NB);
    scan_add   <<<NB, SCAN_B, 0, stream>>>(rowst, bsums, N_NODES);
    fill_csr   <<<cdiv(EN, TB), TB, 0, stream>>>(dst, rowst, cursor, eid,
                                                 N_EDGES, N_NODES);

    const int gemmGrid = cdiv(N_NODES / 16, TB / 32);   // 16x64 strip per wave
    const int aggGrid  = cdiv(N_NODES, TB / 32);        // 1 node per wave

    // ---- layer 1: t = x @ W1 ; hA = relu(gather(t) + b1) ----
    gemm_wmma_strip<<<gemmGrid, TB, 0, stream>>>(x, W1, t, N_NODES, F_IN);
    aggregate_csr  <<<aggGrid, TB, 0, stream>>>(t, src, norm, rowst, counts, eid,
                                                b1, hA, N_EDGES, N_NODES, 1);
    // ---- layer 2 ----
    gemm_wmma_strip<<<gemmGrid, TB, 0, stream>>>(hA, W2, t, N_NODES, H_DIM);
    aggregate_csr  <<<aggGrid, TB, 0, stream>>>(t, src, norm, rowst, counts, eid,
                                                b2, hB, N_EDGES, N_NODES, 1);
    // ---- layer 3 (no relu) ----
    gemm_wmma_strip<<<gemmGrid, TB, 0, stream>>>(hB, W3, t, N_NODES, H_DIM);
    aggregate_csr  <<<aggGrid, TB, 0, stream>>>(t, src, norm, rowst, counts, eid,
                                                b3, hA, N_EDGES, N_NODES, 0);

    // ---- mean pool + final linear ----
    fill_f32   <<<cdiv(G_NUM * H_DIM + G_NUM, TB), TB, 0, stream>>>(
                    sums, 0.0f, G_NUM * H_DIM + G_NUM);
    pool_kernel<<<cdiv(NH, TB), TB, 0, stream>>>(hA, bat, sums, cnt, N_NODES);
    final_fc   <<<cdiv(G_NUM * C_CLS, TB), TB, 0, stream>>>(sums, cnt, Wl, bl, out);
}